// LayerStack_11630771438081
// MI455X (gfx1250) — compile-verified
//
#include <hip/hip_runtime.h>
#include <math.h>

typedef __attribute__((ext_vector_type(2))) float v2f;
typedef __attribute__((ext_vector_type(8))) float v8f;

#define KDIM 128
#define NONE 2048
#define NTWO 64
#define LDA  132   // padded LDS row stride (132 % 64 = 4 banks -> conflict free column walks)

// ---------------------------------------------------------------------------
// Kernel 1: two_mean[i][k] = mean_j emb[two_hop[i][j]][k]   (the 64MB gather)
// ---------------------------------------------------------------------------
__global__ void two_mean_kernel(const float* __restrict__ emb,
                                const int* __restrict__ two_hop,
                                float* __restrict__ two_mean) {
  __shared__ int idx[NTWO];
  const int i = blockIdx.x;
  const int t = threadIdx.x;                 // 0..127 == column k
  if (t < NTWO) idx[t] = two_hop[i * NTWO + t];
  __syncthreads();
  float acc = 0.0f;
  #pragma unroll 4
  for (int j = 0; j < NTWO; ++j) {
    if (j + 1 < NTWO) __builtin_prefetch(&emb[(size_t)idx[j + 1] * KDIM + t], 0, 0);
    acc += emb[(size_t)idx[j] * KDIM + t];   // 512B coalesced per row
  }
  two_mean[(size_t)i * KDIM + t] = acc * (1.0f / NTWO);
}

// ---------------------------------------------------------------------------
// Kernel 2: H = relu(one_feats@Ws^T + two_mean@Wc^T + bc); row-L2-norm.
// 128 blocks x 256 threads. Block = 16-row stripe, wave w = 16-col tile.
// WMMA f32 16x16x4, K swept 4 at a time (32 steps, 2 wmma each).
// ---------------------------------------------------------------------------
__global__ void gemm2h_kernel(const float* __restrict__ emb,
                              const int* __restrict__ one_hop,
                              const float* __restrict__ two_mean,
                              const float* __restrict__ Ws,
                              const float* __restrict__ Wc,
                              const float* __restrict__ bc,
                              float* __restrict__ ohf) {
  __shared__ float A1[16 * LDA];
  __shared__ float A2[16 * LDA];
  __shared__ float Ht[16 * LDA];
  __shared__ float red[256];
  __shared__ float norminv[16];

  const int t  = threadIdx.x;
  const int i0 = blockIdx.x << 4;            // first row of the stripe

  // Stage A tiles: gather one_feats rows + copy two_mean rows into LDS.
  for (int e = t; e < 16 * KDIM; e += 256) {
    const int r = e >> 7, c = e & 127;
    A1[r * LDA + c] = emb[(size_t)one_hop[i0 + r] * KDIM + c];
    A2[r * LDA + c] = two_mean[(size_t)(i0 + r) * KDIM + c];
  }
  __syncthreads();

  const int lane = t & 31;
  const int wave = t >> 5;                   // 0..7
  const int m    = lane & 15;                // tile row (A) / tile col (B,C,D)
  const int kb   = (lane >> 4) << 1;         // K sub-offset 0 or 2 (ISA A/B layout)
  const int n0   = wave << 4;                // column base of this wave's tile

  v8f accS = {0.f,0.f,0.f,0.f,0.f,0.f,0.f,0.f};
  v8f accC = {0.f,0.f,0.f,0.f,0.f,0.f,0.f,0.f};
  const float* wsr = Ws + (size_t)(n0 + m) * KDIM;   // B[k][n] = W[n][k]
  const float* wcr = Wc + (size_t)(n0 + m) * KDIM;

  #pragma unroll 4
  for (int k0 = 0; k0 < KDIM; k0 += 4) {
    const v2f a1 = *(const v2f*)&A1[m * LDA + k0 + kb];
    const v2f b1 = *(const v2f*)&wsr[k0 + kb];
    accS = __builtin_amdgcn_wmma_f32_16x16x4_f32(false, a1, false, b1,
                                                 (short)0, accS, false, false);
    const v2f a2 = *(const v2f*)&A2[m * LDA + k0 + kb];
    const v2f b2 = *(const v2f*)&wcr[k0 + kb];
    accC = __builtin_amdgcn_wmma_f32_16x16x4_f32(false, a2, false, b2,
                                                 (short)0, accC, false, false);
  }

  // D layout: vgpr r, lane -> row = r + 8*(lane>=16), col = n0 + (lane&15)
  const int hi   = lane >> 4;
  const float bias = bc[n0 + m];
  #pragma unroll
  for (int r = 0; r < 8; ++r) {
    const int row = r + (hi << 3);
    const float v = fmaxf(accS[r] + accC[r] + bias, 0.0f);
    Ht[row * LDA + n0 + m] = v;
  }
  __syncthreads();

  // Row-wise sum of squares (fixed order -> deterministic).
  const int row = t >> 4;
  const int c16 = t & 15;
  float ssq = 0.0f;
  #pragma unroll
  for (int q = 0; q < 8; ++q) {
    const float v = Ht[row * LDA + c16 + (q << 4)];
    ssq += v * v;
  }
  red[t] = ssq;
  __syncthreads();
  if (t < 16) {
    float s = 0.0f;
    #pragma unroll
    for (int q = 0; q < 16; ++q) s += red[t * 16 + q];
    norminv[t] = 1.0f / fmaxf(sqrtf(s), 1e-12f);
  }
  __syncthreads();
  const float inv = norminv[row];
  #pragma unroll
  for (int q = 0; q < 8; ++q) {
    const int c = c16 + (q << 4);
    ohf[(size_t)(i0 + row) * KDIM + c] = Ht[row * LDA + c] * inv;
  }
}

// ---------------------------------------------------------------------------
// Kernel 3: partial column sums over one_hop_features (stage 1 of mean).
// ---------------------------------------------------------------------------
__global__ void colsum_kernel(const float* __restrict__ ohf,
                              float* __restrict__ colpart) {
  const int b = blockIdx.x;                  // 0..63, 32 rows each
  const int t = threadIdx.x;                 // column
  float s = 0.0f;
  #pragma unroll 4
  for (int r = 0; r < 32; ++r) s += ohf[(size_t)(b * 32 + r) * KDIM + t];
  colpart[b * KDIM + t] = s;
}

// ---------------------------------------------------------------------------
// Kernel 4: one_mean, ego conv, attention softmax (axis=j), final L2 norm.
// Single block of 128 threads; thread t is row j=t for dW_T and column k=t
// for the softmax / output (so dW_T[k] stays in a register).
// ---------------------------------------------------------------------------
__global__ void final_kernel(const float* __restrict__ emb,
                             const int* __restrict__ v,
                             const float* __restrict__ colpart,
                             const float* __restrict__ Ws1,
                             const float* __restrict__ Wc1,
                             const float* __restrict__ bc1,
                             const float* __restrict__ w_d,
                             const float* __restrict__ w_W,
                             const float* __restrict__ W_mat,
                             float* __restrict__ out) {
  __shared__ float xe[KDIM], om[KDIM], dvec[KDIM], red[KDIM];
  const int t = threadIdx.x;

  // one_mean (stage 2) + ego embedding
  float s = 0.0f;
  #pragma unroll 4
  for (int b = 0; b < 64; ++b) s += colpart[b * KDIM + t];
  om[t] = s * (1.0f / NONE);
  xe[t] = emb[(size_t)v[0] * KDIM + t];
  __syncthreads();

  // h[t] = x@Ws1^T + one_mean@Wc1^T + bc1 ; relu
  float h = bc1[t];
  const float* wsr = Ws1 + (size_t)t * KDIM;
  const float* wcr = Wc1 + (size_t)t * KDIM;
  #pragma unroll 4
  for (int k = 0; k < KDIM; ++k) h += xe[k] * wsr[k] + om[k] * wcr[k];
  h = fmaxf(h, 0.0f);

  // L2 norm of h
  red[t] = h * h;
  __syncthreads();
  for (int off = 64; off > 0; off >>= 1) {
    if (t < off) red[t] += red[t + off];
    __syncthreads();
  }
  const float xk = h * (1.0f / fmaxf(sqrtf(red[0]), 1e-12f));
  __syncthreads();
  dvec[t] = w_d[0] * xk;                     // d = w_d * x
  __syncthreads();

  // dW_T[t] = sum_k d[k] * (w_W * W_mat[t][k])   (thread t owns row j=t)
  float sdw = 0.0f;
  const float* wmr = W_mat + (size_t)t * KDIM;
  #pragma unroll 4
  for (int k = 0; k < KDIM; ++k) sdw += dvec[k] * wmr[k];
  const float dk = sdw * w_W[0];             // == dW_T[k=t]

  // Column-k softmax over j: L[j] = dk * W_mat[j][k]; reads coalesced over k.
  float mx = -INFINITY;
  for (int j = 0; j < KDIM; ++j)
    mx = fmaxf(mx, dk * W_mat[(size_t)j * KDIM + t]);
  float s1 = 0.0f, s2 = 0.0f;
  for (int j = 0; j < KDIM; ++j) {
    const float e = expf(dk * W_mat[(size_t)j * KDIM + t] - mx);
    s1 += e;
    s2 += e * e;                             // Σ e^2 -> column norm without 4th pass
  }
  // out[j][k] = x[k]*att[j][k];  ||out[:,k]|| = |x[k]| * sqrt(Σe²)/Σe
  const float nrm   = fabsf(xk) * sqrtf(s2) / s1;
  const float invn  = 1.0f / fmaxf(nrm, 1e-12f);
  const float scale = (xk / s1) * invn;
  for (int j = 0; j < KDIM; ++j)
    out[(size_t)j * KDIM + t] = scale * expf(dk * W_mat[(size_t)j * KDIM + t] - mx);
}

// ---------------------------------------------------------------------------
// Launch. Workspace layout (floats): two_mean[2048*128] | ohf[2048*128] |
// colpart[64*128]  -> ~2.03 MB total.
// ---------------------------------------------------------------------------
extern "C" void kernel_launch(void* const* d_in, const int* in_sizes, int n_in,
                              void* d_out, int out_size, void* d_ws, size_t ws_size,
                              hipStream_t stream) {
  const float* emb   = (const float*)d_in[0];
  const float* Wc2h  = (const float*)d_in[1];
  const float* bc2h  = (const float*)d_in[2];
  const float* Ws2h  = (const float*)d_in[3];
  const float* Wc1h  = (const float*)d_in[4];
  const float* bc1h  = (const float*)d_in[5];
  const float* Ws1h  = (const float*)d_in[6];
  const float* w_d   = (const float*)d_in[7];
  const float* w_W   = (const float*)d_in[8];
  const float* W_mat = (const float*)d_in[9];
  const int*   v       = (const int*)d_in[10];
  const int*   one_hop = (const int*)d_in[11];
  const int*   two_hop = (const int*)d_in[12];

  float* ws       = (float*)d_ws;
  float* two_mean = ws;                               // 2048*128
  float* ohf      = ws + (size_t)NONE * KDIM;         // 2048*128
  float* colpart  = ws + (size_t)2 * NONE * KDIM;     // 64*128

  two_mean_kernel<<<NONE, KDIM, 0, stream>>>(emb, two_hop, two_mean);
  gemm2h_kernel<<<NONE / 16, 256, 0, stream>>>(emb, one_hop, two_mean,
                                               Ws2h, Wc2h, bc2h, ohf);
  colsum_kernel<<<64, KDIM, 0, stream>>>(ohf, colpart);
  final_kernel<<<1, KDIM, 0, stream>>>(emb, v, colpart, Ws1h, Wc1h, bc1h,
                                       w_d, w_W, W_mat, (float*)d_out);
}